// YOLOv5_14113262535278
// MI455X (gfx1250) — compile-verified
//
#include <hip/hip_runtime.h>

namespace {
constexpr int   kStride = 85;
constexpr int   kNumCls = 80;
constexpr int   kMaxDet = 300;
constexpr float kConfT  = 0.25f;
constexpr float kIouT   = 0.45f;
constexpr float kMaxWH  = 4096.0f;
constexpr float kNegInf = -3.0e38f;
}

// ---------------------------------------------------------------------------
// Phase 1: per-box preprocessing. One thread per box.
//   conf/cls from max/argmax of (cls_score * obj)  (first-occurrence ties),
//   xywh -> xyxy, class-offset boxes (rounded f32, exactly as JAX stores them),
//   raw boxes kept separately for output, score = valid ? conf : -1.
// ---------------------------------------------------------------------------
__global__ __launch_bounds__(256) void yolo_prep(
    const float* __restrict__ pred,
    float4* __restrict__ off4, float4* __restrict__ raw4,
    float* __restrict__ score, float* __restrict__ clsf, int total)
{
    int i = blockIdx.x * blockDim.x + threadIdx.x;
    if (i >= total) return;
    const float* p = pred + (size_t)i * kStride;
    __builtin_prefetch(p, 0, 3);          // global_prefetch_b8 on gfx1250
    float cx = p[0], cy = p[1], w = p[2], h = p[3], obj = p[4];

    float best = -1.0f; int bc = 0;
    #pragma unroll 8
    for (int c = 0; c < kNumCls; ++c) {
        float s = p[5 + c] * obj;         // product order matches reference
        if (s > best) { best = s; bc = c; }   // strict > == first occurrence
    }
    float x1 = cx - w * 0.5f, y1 = cy - h * 0.5f;   // /2 == *0.5 (exact)
    float x2 = cx + w * 0.5f, y2 = cy + h * 0.5f;
    bool valid = (obj > kConfT) && (best > kConfT);
    float fc  = (float)bc;
    float off = fc * kMaxWH;              // exact (int * 2^12)
    off4[i]  = make_float4(x1 + off, y1 + off, x2 + off, y2 + off);
    raw4[i]  = make_float4(x1, y1, x2, y2);
    score[i] = valid ? best : -1.0f;
    clsf[i]  = fc;
}

__device__ __forceinline__ void argmax_comb(float& v, int& i, float ov, int oi) {
    if (ov > v || (ov == v && oi < i)) { v = ov; i = oi; }
}

// ---------------------------------------------------------------------------
// Phase 2: greedy NMS, one workgroup (32 waves) per image.
//   LDS: scores[N] + per-32-chunk maxima[C] + reduction scratch.
//   Each step: argmax over chunk maxima (~N/32 work), in-chunk argmax by
//   wave 0, then a fused suppression + chunk-max-rebuild sweep.
// ---------------------------------------------------------------------------
__global__ __launch_bounds__(1024) void yolo_nms(
    const float4* __restrict__ off4, const float4* __restrict__ raw4,
    const float* __restrict__ score, const float* __restrict__ clsf,
    float* __restrict__ dets, float* __restrict__ keeps, int N)
{
    extern __shared__ float smem[];
    const int C = (N + 31) >> 5;
    float* s_score = smem;                  // N
    float* s_cmax  = smem + N;              // C
    float* s_rv    = s_cmax + C;            // 32
    int*   s_ri    = (int*)(s_rv + 32);     // 32
    float* s_sel   = (float*)(s_ri + 32);   // x1,y1,x2,y2,area,flag

    const int tid = threadIdx.x, nt = blockDim.x;
    const int lane = tid & 31, wid = tid >> 5, nw = nt >> 5;
    const size_t base = (size_t)blockIdx.x * N;
    const float4* boff = off4 + base;
    float* detb  = dets  + (size_t)blockIdx.x * kMaxDet * 6;
    float* keepb = keeps + (size_t)blockIdx.x * kMaxDet;

    // ---- stage scores into LDS: CDNA5 async global->LDS DMA if available ----
#if defined(__gfx1250__) && __has_builtin(__builtin_amdgcn_global_load_async_to_lds_b32) && __has_builtin(__builtin_amdgcn_s_wait_asynccnt)
    for (int i = tid; i < N; i += nt) {
        __builtin_amdgcn_global_load_async_to_lds_b32(
            (__attribute__((address_space(1))) int*)(score + base + i),
            (__attribute__((address_space(3))) int*)(s_score + i), 0, 0);
    }
    __builtin_amdgcn_s_wait_asynccnt(0);
#else
    for (int i = tid; i < N; i += nt) s_score[i] = score[base + i];
#endif
    __syncthreads();

    // ---- initial chunk maxima ----
    for (int c = wid; c < C; c += nw) {
        int i = (c << 5) + lane;
        float v = (i < N) ? s_score[i] : kNegInf;
        for (int m = 16; m > 0; m >>= 1) v = fmaxf(v, __shfl_xor(v, m, 32));
        if (lane == 0) s_cmax[c] = v;
    }
    __syncthreads();

    for (int t = 0; t < kMaxDet; ++t) {
        // ---- argmax over chunk maxima (ties -> lower chunk == lower index) ----
        float bv = kNegInf; int bi = 0x7fffffff;
        for (int c = tid; c < C; c += nt) argmax_comb(bv, bi, s_cmax[c], c);
        for (int m = 16; m > 0; m >>= 1)
            argmax_comb(bv, bi, __shfl_xor(bv, m, 32), __shfl_xor(bi, m, 32));
        if (lane == 0) { s_rv[wid] = bv; s_ri[wid] = bi; }
        __syncthreads();

        if (wid == 0) {
            float cv = (lane < nw) ? s_rv[lane] : kNegInf;
            int   ci = (lane < nw) ? s_ri[lane] : 0x7fffffff;
            for (int m = 16; m > 0; m >>= 1)
                argmax_comb(cv, ci, __shfl_xor(cv, m, 32), __shfl_xor(ci, m, 32));
            // butterfly: every lane of wave 0 holds winner (cv, ci)
            if (cv > 0.0f) {
                int   myi = (ci << 5) + lane;
                float myv = (myi < N) ? s_score[myi] : kNegInf;
                int   idx = (myi < N) ? myi : 0x7fffffff;
                float rv = myv; int ri = idx;
                for (int m = 16; m > 0; m >>= 1)
                    argmax_comb(rv, ri, __shfl_xor(rv, m, 32), __shfl_xor(ri, m, 32));
                float v2 = (idx == ri) ? -1.0f : myv;       // chunk max sans winner
                for (int m = 16; m > 0; m >>= 1) v2 = fmaxf(v2, __shfl_xor(v2, m, 32));
                if (lane == 0) {
                    float4 ob = boff[ri];
                    float4 rb = raw4[base + ri];
                    float  fc = clsf[base + ri];
                    float* d = detb + t * 6;
                    d[0] = rb.x; d[1] = rb.y; d[2] = rb.z; d[3] = rb.w;
                    d[4] = rv;   d[5] = fc;
                    keepb[t]    = 1.0f;
                    s_score[ri] = -1.0f;
                    s_cmax[ci]  = v2;
                    s_sel[0] = ob.x; s_sel[1] = ob.y; s_sel[2] = ob.z; s_sel[3] = ob.w;
                    s_sel[4] = (ob.z - ob.x) * (ob.w - ob.y);
                    s_sel[5] = 1.0f;
                }
            } else if (lane == 0) {
                s_sel[5] = 0.0f;
            }
        }
        __syncthreads();

        if (s_sel[5] == 0.0f) {           // nothing live: remaining rows are zero
            for (int j = tid; j < (kMaxDet - t) * 6; j += nt) detb[t * 6 + j] = 0.0f;
            for (int j = t + tid; j < kMaxDet; j += nt)       keepb[j] = 0.0f;
            break;
        }

        // ---- fused suppression + chunk-max rebuild sweep ----
        const float sx1 = s_sel[0], sy1 = s_sel[1], sx2 = s_sel[2], sy2 = s_sel[3];
        const float sa  = s_sel[4];
        for (int c = wid; c < C; c += nw) {
            int i = (c << 5) + lane;
            float v = (i < N) ? s_score[i] : kNegInf;
            if (i < N && v > 0.0f) {      // dead boxes: skip L2 box read
                float4 bb = boff[i];
                float lx = fmaxf(sx1, bb.x), ly = fmaxf(sy1, bb.y);
                float rx = fminf(sx2, bb.z), ry = fminf(sy2, bb.w);
                float iw = fmaxf(rx - lx, 0.0f), ih = fmaxf(ry - ly, 0.0f);
                float inter = iw * ih;
                float a2 = (bb.z - bb.x) * (bb.w - bb.y);
                float iou = inter / (sa + a2 - inter);
                if (iou > kIouT) { v = -1.0f; s_score[i] = v; }
            }
            for (int m = 16; m > 0; m >>= 1) v = fmaxf(v, __shfl_xor(v, m, 32));
            if (lane == 0) s_cmax[c] = v;
        }
        __syncthreads();
    }
}

extern "C" void kernel_launch(void* const* d_in, const int* in_sizes, int n_in,
                              void* d_out, int out_size, void* d_ws, size_t ws_size,
                              hipStream_t stream) {
    (void)n_in; (void)ws_size;
    const float* pred = (const float*)d_in[0];
    const int B     = out_size / (kMaxDet * 7);   // dets(6) + keeps(1) per slot
    const int total = in_sizes[0] / kStride;      // B*N
    const int N     = total / B;

    float*  ws    = (float*)d_ws;                 // 10 floats per box ~ 16.1 MB
    float4* off4  = (float4*)ws;
    float4* raw4  = (float4*)(ws + (size_t)total * 4);
    float*  score = ws + (size_t)total * 8;
    float*  clsf  = ws + (size_t)total * 9;

    float* dets  = (float*)d_out;
    float* keeps = dets + (size_t)B * kMaxDet * 6;

    yolo_prep<<<dim3((total + 255) / 256), dim3(256), 0, stream>>>(
        pred, off4, raw4, score, clsf, total);

    const int C = (N + 31) >> 5;
    size_t lds = ((size_t)N + C + 32 + 32 + 8) * sizeof(float);
    yolo_nms<<<dim3(B), dim3(1024), lds, stream>>>(
        off4, raw4, score, clsf, dets, keeps, N);
}